// SummarizeLayer_64493228917137
// MI455X (gfx1250) — compile-verified
//
#include <hip/hip_runtime.h>
#include <hip/hip_bf16.h>
#include <math.h>

typedef float v2f __attribute__((ext_vector_type(2)));
typedef float v8f __attribute__((ext_vector_type(8)));
typedef unsigned u32x4 __attribute__((ext_vector_type(4)));
typedef int i32x4 __attribute__((ext_vector_type(4)));
typedef int i32x8 __attribute__((ext_vector_type(8)));

#define BATCH 4
#define NROW  32768
#define FDIM  256
#define KSEL  4096

// ---- order-preserving float<->u32 key transform ----
__device__ __forceinline__ unsigned f2k(float f) {
  unsigned u = __float_as_uint(f);
  return (u & 0x80000000u) ? ~u : (u | 0x80000000u);
}
__device__ __forceinline__ float k2f(unsigned k) {
  unsigned u = (k & 0x80000000u) ? (k & 0x7FFFFFFFu) : ~k;
  return __uint_as_float(u);
}

// ---- K1: w = p / ||p|| ----
__global__ void kw_kernel(const float* __restrict__ p, float* __restrict__ w) {
  __shared__ float s[256];
  int t = threadIdx.x;
  float v = p[t];
  s[t] = v * v;
  __syncthreads();
  for (int o = 128; o > 0; o >>= 1) {
    if (t < o) s[t] += s[t + o];
    __syncthreads();
  }
  w[t] = v * rsqrtf(s[0]);
}

// ---- K2: y[b,n] = x[b,n,:].w  via V_WMMA_F32_16X16X4_F32 ----
// w (1x256 f32 tile) is staged into LDS by the Tensor Data Mover (one
// TENSOR_LOAD_TO_LDS issued by wave 0), completion via s_wait_tensorcnt.
// Each wave then computes 16 rows: A(16x4) per doc layout (lanes 0-15 hold
// K0/K1 in VGPR0/1, lanes 16-31 hold K2/K3); B(4x16) broadcast so every D
// column accumulates y. Lane 0 ends with y[rows 0..7] in acc[0..7], lane 16
// with y[rows 8..15].
__global__ void kmatvec_kernel(const float* __restrict__ x,
                               const float* __restrict__ w,
                               unsigned* __restrict__ yk) {
  __shared__ float ws[FDIM];
  int tid = threadIdx.x;

  if (tid < 32) {  // wave 0 issues one TDM DMA: global w[0..255] -> LDS ws
    unsigned ldsOff = (unsigned)(uintptr_t)&ws[0];       // low 32b = LDS offset
    unsigned long long ga = (unsigned long long)(uintptr_t)w;
    u32x4 g0;
    g0.x = 1u;                                            // count=1, user mode
    g0.y = ldsOff;                                        // lds_addr
    g0.z = (unsigned)ga;                                  // global_addr[31:0]
    g0.w = (unsigned)((ga >> 32) & 0x01FFFFFFu) | (2u << 30);  // addr[56:32], type=2
    i32x8 g1;
    g1[0] = (int)(2u << 16);        // workgroup_mask=0, data_size=2 (4 bytes)
    g1[1] = (int)(256u << 16);      // tensor_dim0[15:0]=256 (atomic_addr=0)
    g1[2] = (int)(1u << 16);        // tensor_dim0[31:16]=0, tensor_dim1[15:0]=1
    g1[3] = (int)(256u << 16);      // tensor_dim1[31:16]=0, tile_dim0=256
    g1[4] = 1;                      // tile_dim1=1, tile_dim2=0
    g1[5] = 256;                    // tensor_dim0_stride[31:0]=256
    g1[6] = 0;                      // stride hi / dim1_stride lo
    g1[7] = 0;
    i32x4 gz = {0, 0, 0, 0};        // groups 2,3: higher dims unused
    i32x8 gz8 = {0, 0, 0, 0, 0, 0, 0, 0};
    __builtin_amdgcn_tensor_load_to_lds(g0, g1, gz, gz, gz8, 0);
  }
  __builtin_amdgcn_s_wait_tensorcnt(0);  // wave0 waits its DMA; others pass
  __syncthreads();

  int gw   = blockIdx.x * 8 + (tid >> 5);  // global wave id, 8192 total
  int lane = tid & 31;
  int rowBase = gw * 16;
  int rowA = rowBase + (lane & 15);
  int ksel = (lane < 16) ? 0 : 2;
  const float* xrow = x + (size_t)rowA * FDIM + ksel;

  v8f acc = {};
  #pragma unroll 4
  for (int c = 0; c < 64; ++c) {
    v2f a = *(const v2f*)(xrow + c * 4);      // x[rowA, 4c+ksel .. +1]
    v2f b;
    b.x = ws[4 * c + ksel];
    b.y = ws[4 * c + ksel + 1];
    acc = __builtin_amdgcn_wmma_f32_16x16x4_f32(
        /*neg_a=*/false, a, /*neg_b=*/false, b,
        /*c_mod=*/(short)0, acc, /*reuse_a=*/false, /*reuse_b=*/false);
  }

  if (lane == 0 || lane == 16) {
    int base = rowBase + ((lane == 16) ? 8 : 0);
    #pragma unroll
    for (int r = 0; r < 8; ++r) yk[base + r] = f2k(acc[r]);
  }
}

// ---- K3: histogram of high 16 bits of keys, per batch ----
__global__ void khist_hi_kernel(const unsigned* __restrict__ yk,
                                unsigned* __restrict__ histHi) {
  int g = blockIdx.x * blockDim.x + threadIdx.x;  // 131072 threads
  int b = g >> 15;
  unsigned key = yk[g];
  atomicAdd(&histHi[(b << 16) + (key >> 16)], 1u);
}

// scal layout (u32): [0..3]=H  [4..7]=CgtHi  [8..11]=remaining
//                    [12..15]=T [16..19]=Cgt [20..23]=tiesNeeded
//                    [24..27]=ctrGT [28..31]=ctrTie
__global__ void kfind_hi_kernel(const unsigned* __restrict__ histHi,
                                unsigned* __restrict__ scal) {
  int b = threadIdx.x;  // 0..3
  const unsigned* h = histHi + (b << 16);
  unsigned cum = 0;
  for (int bucket = 65535; bucket >= 0; --bucket) {
    unsigned c = h[bucket];
    if (cum + c >= (unsigned)KSEL) {
      scal[0 + b]  = (unsigned)bucket;
      scal[4 + b]  = cum;
      scal[8 + b]  = (unsigned)KSEL - cum;
      return;
    }
    cum += c;
  }
}

// ---- K5: low-16-bit histogram restricted to boundary bucket ----
__global__ void khist_lo_kernel(const unsigned* __restrict__ yk,
                                const unsigned* __restrict__ scal,
                                unsigned* __restrict__ histLo) {
  int g = blockIdx.x * blockDim.x + threadIdx.x;
  int b = g >> 15;
  unsigned key = yk[g];
  if ((key >> 16) == scal[b])
    atomicAdd(&histLo[(b << 16) + (key & 0xFFFFu)], 1u);
}

__global__ void kfind_lo_kernel(const unsigned* __restrict__ histLo,
                                unsigned* __restrict__ scal) {
  int b = threadIdx.x;
  unsigned H = scal[0 + b], CgtHi = scal[4 + b], rem = scal[8 + b];
  const unsigned* h = histLo + (b << 16);
  unsigned cum = 0;
  for (int bucket = 65535; bucket >= 0; --bucket) {
    unsigned c = h[bucket];
    if (cum + c >= rem) {
      scal[12 + b] = (H << 16) | (unsigned)bucket;  // exact threshold key T
      scal[16 + b] = CgtHi + cum;                   // strictly-greater count
      scal[20 + b] = rem - cum;                     // ties needed (>=1)
      return;
    }
    cum += c;
  }
}

// ---- K7: compact keys > T; collect ties == T ----
__global__ void kcollect_kernel(const unsigned* __restrict__ yk,
                                unsigned* __restrict__ scal,
                                unsigned* __restrict__ selKey,
                                unsigned* __restrict__ selIdx,
                                unsigned* __restrict__ tieBuf) {
  int g = blockIdx.x * blockDim.x + threadIdx.x;
  int b = g >> 15;
  unsigned n = (unsigned)(g & 32767);
  unsigned key = yk[g];
  unsigned T = scal[12 + b];
  if (key > T) {
    unsigned pos = atomicAdd(&scal[24 + b], 1u);  // pos <= Cgt-1 <= 4094
    selKey[(b << 12) + pos] = key;
    selIdx[(b << 12) + pos] = n;
  } else if (key == T) {
    unsigned pos = atomicAdd(&scal[28 + b], 1u);
    if (pos < (unsigned)KSEL) tieBuf[(b << 12) + pos] = n;
  }
}

// ---- K8: per batch: resolve ties (stable by index), bitonic-sort top-K ----
__global__ void __launch_bounds__(1024)
ksort_kernel(const unsigned* __restrict__ scal,
             const unsigned* __restrict__ tieBuf,
             const unsigned* __restrict__ selKey,
             const unsigned* __restrict__ selIdx,
             unsigned* __restrict__ outIdx,
             float* __restrict__ outTanh) {
  __shared__ unsigned tie[KSEL];                 // 16 KB
  __shared__ unsigned long long arr[KSEL];       // 32 KB
  int b = blockIdx.x;
  unsigned tid = threadIdx.x;
  unsigned T = scal[12 + b], Cgt = scal[16 + b];
  unsigned nt = scal[28 + b];
  if (nt > (unsigned)KSEL) nt = (unsigned)KSEL;

  for (unsigned i = tid; i < KSEL; i += 1024)
    tie[i] = (i < nt) ? tieBuf[(b << 12) + i] : 0xFFFFFFFFu;

  // bitonic ascending sort of tie indices (stable tie-break = lowest index)
  for (unsigned kk = 2; kk <= KSEL; kk <<= 1) {
    for (unsigned j = kk >> 1; j > 0; j >>= 1) {
      __syncthreads();
      for (unsigned i = tid; i < KSEL; i += 1024) {
        unsigned ixj = i ^ j;
        if (ixj > i) {
          unsigned a = tie[i], c = tie[ixj];
          bool asc = ((i & kk) == 0);
          if ((a > c) == asc) { tie[i] = c; tie[ixj] = a; }
        }
      }
    }
  }
  __syncthreads();

  // build exactly K entries: (~key)<<32 | idx  -> ascending sort gives
  // descending value, ascending index on ties
  for (unsigned i = tid; i < KSEL; i += 1024) {
    unsigned key, idx;
    if (i < Cgt) { key = selKey[(b << 12) + i]; idx = selIdx[(b << 12) + i]; }
    else         { key = T;                     idx = tie[i - Cgt]; }
    arr[i] = ((unsigned long long)(~key) << 32) | (unsigned long long)idx;
  }

  for (unsigned kk = 2; kk <= KSEL; kk <<= 1) {
    for (unsigned j = kk >> 1; j > 0; j >>= 1) {
      __syncthreads();
      for (unsigned i = tid; i < KSEL; i += 1024) {
        unsigned ixj = i ^ j;
        if (ixj > i) {
          unsigned long long a = arr[i], c = arr[ixj];
          bool asc = ((i & kk) == 0);
          if ((a > c) == asc) { arr[i] = c; arr[ixj] = a; }
        }
      }
    }
  }
  __syncthreads();

  for (unsigned j = tid; j < KSEL; j += 1024) {
    unsigned long long v = arr[j];
    unsigned key = ~(unsigned)(v >> 32);
    outIdx[(b << 12) + j]  = (unsigned)v;
    outTanh[(b << 12) + j] = tanhf(k2f(key));
  }
}

// ---- K10: out[b,i,j,f] = x[b, idx[i,j], f] * tanh(vals[i,j]) ----
__global__ void kscatter_kernel(const float* __restrict__ x,
                                const unsigned* __restrict__ outIdx,
                                const float* __restrict__ outTanh,
                                float* __restrict__ out) {
  int wid  = blockIdx.x * 8 + (threadIdx.x >> 5);  // 65536 rows
  int lane = threadIdx.x & 31;
  int b = wid >> 14;
  int i = (wid >> 12) & 3;
  int j = wid & 4095;
  unsigned idx = outIdx[(i << 12) + j];
  float s      = outTanh[(i << 12) + j];
  const float4* src = (const float4*)(x + ((size_t)b * NROW + idx) * FDIM);
  float4* dst = (float4*)(out + (size_t)wid * FDIM);
  __builtin_prefetch(src, 0, 0);  // global_prefetch_b8: warm gather row
  #pragma unroll
  for (int t = lane; t < FDIM / 4; t += 32) {
    float4 v = src[t];
    v.x *= s; v.y *= s; v.z *= s; v.w *= s;
    dst[t] = v;
  }
}

extern "C" void kernel_launch(void* const* d_in, const int* in_sizes, int n_in,
                              void* d_out, int out_size, void* d_ws, size_t ws_size,
                              hipStream_t stream) {
  const float* x = (const float*)d_in[0];
  const float* p = (const float*)d_in[1];
  (void)in_sizes; (void)n_in; (void)out_size; (void)ws_size;

  char* ws = (char*)d_ws;
  float*    w       = (float*)(ws + 0);             // 1 KB
  unsigned* yk      = (unsigned*)(ws + 1024);       // 512 KB keys
  unsigned* histHi  = (unsigned*)(ws + 525312);     // 1 MB
  unsigned* histLo  = (unsigned*)(ws + 1573888);    // 1 MB
  unsigned* scal    = (unsigned*)(ws + 2622464);    // 256 B
  unsigned* tieBuf  = (unsigned*)(ws + 2622720);    // 64 KB
  unsigned* selKey  = (unsigned*)(ws + 2688256);    // 64 KB
  unsigned* selIdx  = (unsigned*)(ws + 2753792);    // 64 KB
  unsigned* outIdx  = (unsigned*)(ws + 2819328);    // 64 KB
  float*    outTanh = (float*)(ws + 2884864);       // 64 KB
  float* out = (float*)d_out;

  // zero histHi + histLo + scal (contiguous) every call
  (void)hipMemsetAsync(histHi, 0, 2u * 1048576u + 256u, stream);

  kw_kernel     <<<1,    256, 0, stream>>>(p, w);
  kmatvec_kernel<<<1024, 256, 0, stream>>>(x, w, yk);
  khist_hi_kernel<<<512, 256, 0, stream>>>(yk, histHi);
  kfind_hi_kernel<<<1,   4,   0, stream>>>(histHi, scal);
  khist_lo_kernel<<<512, 256, 0, stream>>>(yk, scal, histLo);
  kfind_lo_kernel<<<1,   4,   0, stream>>>(histLo, scal);
  kcollect_kernel<<<512, 256, 0, stream>>>(yk, scal, selKey, selIdx, tieBuf);
  ksort_kernel  <<<4,    1024,0, stream>>>(scal, tieBuf, selKey, selIdx, outIdx, outTanh);
  kscatter_kernel<<<8192,256, 0, stream>>>(x, outIdx, outTanh, out);
}